// DTC_7000796692862
// MI455X (gfx1250) — compile-verified
//
#include <hip/hip_runtime.h>

// Problem constants (match reference)
#define Nn    65536
#define DIN   1024
#define DLAT  128
#define KCENT 512
#define ROWS  64               // rows of x per block in main kernel
#define NBLK  (Nn / ROWS)      // 1024 blocks

// CDNA5 WMMA vector types
typedef __attribute__((ext_vector_type(2))) float v2f;
typedef __attribute__((ext_vector_type(8))) float v8f;

__device__ __forceinline__ v8f wmma_f32(v2f a, v2f b, v8f c) {
  // V_WMMA_F32_16X16X4_F32 : full fp32 matrix op, fp32 accumulate
  return __builtin_amdgcn_wmma_f32_16x16x4_f32(false, a, false, b,
                                               (short)0, c, false, false);
}

// ---- async global->LDS staging (CDNA5 ASYNCcnt path), guarded ----
#if defined(__has_builtin)
#if __has_builtin(__builtin_amdgcn_global_load_async_to_lds_b128) && \
    __has_builtin(__builtin_amdgcn_s_wait_asynccnt)
#define HAVE_ASYNC 1
#endif
#endif

#ifdef HAVE_ASYNC
// builtin signature (from hipcc diagnostic): param0 is
//   'int __attribute__((vector_size(16))) __device__ *'  (global int4*)
typedef __attribute__((vector_size(16))) int vi4;
typedef __attribute__((address_space(1))) vi4 gvi4;   // global
typedef __attribute__((address_space(3))) vi4 lvi4;   // LDS

__device__ __forceinline__ void stage16(const float* g, float* l) {
  // per-lane 16B DMA: GLOBAL_LOAD_ASYNC_TO_LDS_B128
  __builtin_amdgcn_global_load_async_to_lds_b128(
      (gvi4*)(float*)g, (lvi4*)l, 0, 0);
}
__device__ __forceinline__ void async_wait() {
  __builtin_amdgcn_s_wait_asynccnt(0);   // s_wait_asynccnt 0
}
#else
__device__ __forceinline__ void stage16(const float* g, float* l) {
  *(float4*)l = *(const float4*)g;
}
__device__ __forceinline__ void async_wait() {}
#endif

// ---- LDS layout (float offsets), dynamic shared memory ----
// Padded strides: (stride % 64) == 4 -> A/B fragment reads hit banks 4*lane+k,
// distinct across 16 lanes; 16B-aligned rows so b128 transfers are legal.
#define XSTR 68
#define WSTR 128
#define CSTR 132
#define LSTR 132
#define ABUF_FLOATS 12544                    // xbuf 64*68 + wbuf 64*128
#define OFF_XBUF0 0
#define OFF_WBUF0 (ROWS * XSTR)              // 4352
#define UNION_FLOATS (2 * ABUF_FLOATS)       // 25088 (>= centbuf 128*132=16896)
#define OFF_CENT  0
#define OFF_LTILE UNION_FLOATS               // 25088, size 64*132 = 8448
#define OFF_BBUF  (OFF_LTILE + ROWS * LSTR)  // 33536
#define OFF_CNS   (OFF_BBUF + DLAT)          // 33664
#define OFF_LNORM (OFF_CNS + KCENT)          // 34176
#define OFF_RSUM  (OFF_LNORM + ROWS)         // 34240
#define OFF_FQL   (OFF_RSUM + ROWS)          // 34304
#define SMEM_FLOATS (OFF_FQL + KCENT)        // 34816 floats = 136 KiB

// ---------------- K0: centroid squared norms ----------------
__global__ void __launch_bounds__(256) cnorm_kernel(const float* __restrict__ cent,
                                                    float* __restrict__ cnorm) {
  int k = blockIdx.x * 256 + threadIdx.x;
  if (k < KCENT) {
    float s = 0.f;
    const float* cr = cent + (size_t)k * DLAT;
    #pragma unroll 8
    for (int i = 0; i < DLAT; ++i) { float v = cr[i]; s += v * v; }
    cnorm[k] = s;
  }
}

// cooperative stage of one (x 64x64, W 64x128) chunk into buffer `buf`
__device__ __forceinline__ void stage_xw(float* sm, const float* xrow,
                                         const float* W, int kc, int buf, int tid) {
  float* xb = sm + buf * ABUF_FLOATS + OFF_XBUF0;
  float* wb = sm + buf * ABUF_FLOATS + OFF_WBUF0;
  int r0 = tid >> 4, c0 = (tid & 15) << 2;
  #pragma unroll
  for (int p = 0; p < 4; ++p) {
    int r = r0 + p * 16;
    stage16(xrow + (size_t)r * DIN + kc + c0, xb + r * XSTR + c0);
  }
  int wr0 = tid >> 5, wc0 = (tid & 31) << 2;
  #pragma unroll
  for (int p = 0; p < 8; ++p) {
    int r = wr0 + p * 8;
    stage16(W + (size_t)(kc + r) * DLAT + wc0, wb + r * WSTR + wc0);
  }
}

// ---------------- K1: fused  l = xW+b,  Q,  Fq partials ----------------
__global__ void __launch_bounds__(256) dec_main(const float* __restrict__ x,
                                                const float* __restrict__ W,
                                                const float* __restrict__ bias,
                                                const float* __restrict__ cent,
                                                const float* __restrict__ cnorm_g,
                                                float* __restrict__ l_out,
                                                float* __restrict__ q_out,
                                                float* __restrict__ fq_part) {
  extern __shared__ float sm[];
  const int tid   = threadIdx.x;
  const int lane  = tid & 31;
  const int wave  = tid >> 5;
  const int mt    = wave & 3;     // M-tile (4 tiles of 16 rows)
  const int nh    = wave >> 2;    // N half
  const int m0    = mt * 16;
  const int lhalf = lane >> 4;    // A/B: lanes 16..31 carry K+2,K+3; C: rows +8
  const int l15   = lane & 15;
  const int rowbase = blockIdx.x * ROWS;

  // prologue: constants into LDS, zero reductions
  for (int i = tid; i < DLAT;  i += 256) sm[OFF_BBUF + i] = bias[i];
  for (int i = tid; i < KCENT; i += 256) sm[OFF_CNS + i]  = cnorm_g[i];
  for (int i = tid; i < KCENT; i += 256) sm[OFF_FQL + i]  = 0.f;
  if (tid < ROWS) sm[OFF_RSUM + tid] = 0.f;

  // ---------------- Stage A: l = x @ W + b  (double-buffered DMA) ----------------
  v8f accA[4];
  #pragma unroll
  for (int j = 0; j < 4; ++j)
    #pragma unroll
    for (int r = 0; r < 8; ++r) accA[j][r] = 0.f;

  const float* xrow = x + (size_t)rowbase * DIN;
  stage_xw(sm, xrow, W, 0, 0, tid);            // preload chunk 0
  for (int kc = 0; kc < DIN; kc += 64) {
    const int cur = (kc >> 6) & 1;
    async_wait();                              // current buffer resident
    __syncthreads();
    if (kc + 64 < DIN) stage_xw(sm, xrow, W, kc + 64, cur ^ 1, tid);  // overlap DMA
    const float* xb = sm + cur * ABUF_FLOATS + OFF_XBUF0;
    const float* wb = sm + cur * ABUF_FLOATS + OFF_WBUF0;
    #pragma unroll
    for (int ks = 0; ks < 16; ++ks) {
      const int k4 = ks * 4 + lhalf * 2;
      const int arow = m0 + l15;
      v2f a;
      a.x = xb[arow * XSTR + k4];
      a.y = xb[arow * XSTR + k4 + 1];
      #pragma unroll
      for (int j = 0; j < 4; ++j) {
        const int col = nh * 64 + j * 16 + l15;
        v2f bf;
        bf.x = wb[k4 * WSTR + col];
        bf.y = wb[(k4 + 1) * WSTR + col];
        accA[j] = wmma_f32(a, bf, accA[j]);
      }
    }
  }

  // epilogue A: + bias, stash l in LDS and stream to d_out
  #pragma unroll
  for (int j = 0; j < 4; ++j) {
    const int col = nh * 64 + j * 16 + l15;
    const float bb = sm[OFF_BBUF + col];
    #pragma unroll
    for (int r = 0; r < 8; ++r) {
      const int rr = m0 + r + lhalf * 8;
      const float val = accA[j][r] + bb;
      sm[OFF_LTILE + rr * LSTR + col] = val;
      l_out[(size_t)(rowbase + rr) * DLAT + col] = val;
    }
  }
  __syncthreads();
  if (tid < ROWS) {   // ||l_row||^2
    float s = 0.f;
    const float* lr = &sm[OFF_LTILE + tid * LSTR];
    #pragma unroll 8
    for (int k = 0; k < DLAT; ++k) { float v = lr[k]; s += v * v; }
    sm[OFF_LNORM + tid] = s;
  }

  // ---------------- Stage B: dot = l @ centroids^T ----------------
  v8f accB[16];
  #pragma unroll
  for (int t = 0; t < 16; ++t)
    #pragma unroll
    for (int r = 0; r < 8; ++r) accB[t][r] = 0.f;

  #pragma unroll
  for (int c = 0; c < 4; ++c) {          // 4 chunks of 128 centroids
    __syncthreads();
    {   // centroid chunk: 128 rows x 128 feats -> LDS (reuses stage-A union)
      int r0 = tid >> 1, c0 = (tid & 1) << 6;
      #pragma unroll
      for (int p = 0; p < 16; ++p)
        stage16(cent + (size_t)(c * 128 + r0) * DLAT + c0 + p * 4,
                sm + OFF_CENT + r0 * CSTR + c0 + p * 4);
      if (c < 3)   // keep next chunk warm -> global_prefetch_b8
        __builtin_prefetch(cent + (size_t)((c + 1) * 128 + r0) * DLAT + c0, 0, 2);
    }
    async_wait();
    __syncthreads();
    #pragma unroll
    for (int ks = 0; ks < 32; ++ks) {    // K = 128
      const int k4 = ks * 4 + lhalf * 2;
      const int arow = m0 + l15;
      v2f a;
      a.x = sm[OFF_LTILE + arow * LSTR + k4];
      a.y = sm[OFF_LTILE + arow * LSTR + k4 + 1];
      #pragma unroll
      for (int j = 0; j < 4; ++j) {
        const int n = nh * 64 + j * 16 + l15;   // B[k][n] = centroids[n][k]
        v2f bf;
        bf.x = sm[OFF_CENT + n * CSTR + k4];
        bf.y = sm[OFF_CENT + n * CSTR + k4 + 1];
        accB[c * 4 + j] = wmma_f32(a, bf, accB[c * 4 + j]);
      }
    }
  }

  // ---------------- Epilogue: num = 1/(1+D), Q, Fq partials ----------------
  const int rbase = m0 + lhalf * 8;
  float lnorm_r[8], rowpart[8];
  #pragma unroll
  for (int r = 0; r < 8; ++r) { lnorm_r[r] = sm[OFF_LNORM + rbase + r]; rowpart[r] = 0.f; }

  #pragma unroll
  for (int t = 0; t < 16; ++t) {
    const int cbase = (t >> 2) * 128 + nh * 64 + (t & 3) * 16;
    const float cn = sm[OFF_CNS + cbase + l15];
    #pragma unroll
    for (int r = 0; r < 8; ++r) {
      const float Dv   = lnorm_r[r] + cn - 2.f * accB[t][r];
      const float numv = 1.f / (1.f + Dv);          // alpha=1 => power -1
      accB[t][r] = numv;
      rowpart[r] += numv;
    }
  }
  #pragma unroll
  for (int r = 0; r < 8; ++r) {         // reduce over 16-lane groups
    float v = rowpart[r];
    v += __shfl_xor(v, 1, 32);
    v += __shfl_xor(v, 2, 32);
    v += __shfl_xor(v, 4, 32);
    v += __shfl_xor(v, 8, 32);
    if (l15 == 0) atomicAdd(&sm[OFF_RSUM + rbase + r], v);   // ds_add_f32
  }
  __syncthreads();
  float invr[8];
  #pragma unroll
  for (int r = 0; r < 8; ++r) invr[r] = 1.f / sm[OFF_RSUM + rbase + r];

  #pragma unroll
  for (int t = 0; t < 16; ++t) {
    const int cbase = (t >> 2) * 128 + nh * 64 + (t & 3) * 16;
    float colpart = 0.f;
    #pragma unroll
    for (int r = 0; r < 8; ++r) {
      const float q = accB[t][r] * invr[r];
      q_out[(size_t)(rowbase + rbase + r) * KCENT + cbase + l15] = q;
      colpart += q;
    }
    colpart += __shfl_xor(colpart, 16, 32);   // combine the two row-halves
    if (lane < 16) atomicAdd(&sm[OFF_FQL + cbase + lane], colpart);
  }
  __syncthreads();
  float* dst = fq_part + (size_t)blockIdx.x * KCENT;
  for (int i = tid; i < KCENT; i += 256) dst[i] = sm[OFF_FQL + i];
}

// ---------------- K2: reduce Fq partials over blocks ----------------
__global__ void __launch_bounds__(256) fq_reduce(const float* __restrict__ parts,
                                                 float* __restrict__ Fq) {
  int k = blockIdx.x * 256 + threadIdx.x;
  if (k < KCENT) {
    float s = 0.f;
    for (int b = 0; b < NBLK; ++b) s += parts[(size_t)b * KCENT + k];
    Fq[k] = s;
  }
}

// ---------------- K3: P = (Q^2/Fq) row-normalized ----------------
__global__ void __launch_bounds__(256) p_kernel(const float* __restrict__ Q,
                                                const float* __restrict__ Fq,
                                                float* __restrict__ P) {
  __shared__ float invfq[KCENT];
  const int tid = threadIdx.x;
  for (int i = tid; i < KCENT; i += 256) invfq[i] = 1.f / Fq[i];
  __syncthreads();
  const int lane = tid & 31;
  const int wave = tid >> 5;
  const size_t row = (size_t)blockIdx.x * 8 + wave;   // one wave per row
  const float* qr = Q + row * KCENT;
  float vals[16], s = 0.f;
  #pragma unroll
  for (int i = 0; i < 16; ++i) {
    const int col = lane + 32 * i;
    const float q = qr[col];
    const float n2 = q * q * invfq[col];
    vals[i] = n2;
    s += n2;
  }
  s += __shfl_xor(s, 1, 32);
  s += __shfl_xor(s, 2, 32);
  s += __shfl_xor(s, 4, 32);
  s += __shfl_xor(s, 8, 32);
  s += __shfl_xor(s, 16, 32);
  const float inv = 1.f / s;
  float* pr = P + row * KCENT;
  #pragma unroll
  for (int i = 0; i < 16; ++i) pr[lane + 32 * i] = vals[i] * inv;
}

// ---------------- launcher ----------------
extern "C" void kernel_launch(void* const* d_in, const int* in_sizes, int n_in,
                              void* d_out, int out_size, void* d_ws, size_t ws_size,
                              hipStream_t stream) {
  const float* x    = (const float*)d_in[0];
  const float* W    = (const float*)d_in[1];
  const float* b    = (const float*)d_in[2];
  const float* cent = (const float*)d_in[3];

  float* l_out = (float*)d_out;                    // [N, 128]
  float* q_out = l_out + (size_t)Nn * DLAT;        // [N, 512]
  float* p_out = q_out + (size_t)Nn * KCENT;       // [N, 512]

  float* wsf   = (float*)d_ws;
  float* cnorm = wsf;              // 512
  float* Fq    = wsf + 512;        // 512
  float* parts = wsf + 1024;       // NBLK*512 (2 MiB)

  cnorm_kernel<<<2, 256, 0, stream>>>(cent, cnorm);
  const size_t smem = SMEM_FLOATS * sizeof(float); // 136 KiB dynamic LDS
  dec_main<<<NBLK, 256, smem, stream>>>(x, W, b, cent, cnorm, l_out, q_out, parts);
  fq_reduce<<<2, 256, 0, stream>>>(parts, Fq);
  p_kernel<<<Nn / 8, 256, 0, stream>>>(q_out, Fq, p_out);
}